// VQSign_84885733638452
// MI455X (gfx1250) — compile-verified
//
#include <hip/hip_runtime.h>

typedef __attribute__((ext_vector_type(2))) float v2f;
typedef __attribute__((ext_vector_type(8))) float v8f;

#define B_  64
#define T_  512
#define D_  512
#define K_  256
#define G3  (3*D_)

// -----------------------------------------------------------------------------
// Async global->LDS copy (CDNA5 GLOBAL_LOAD_ASYNC_TO_LDS_B128, ASYNCcnt-tracked).
// The low 32 bits of a generic pointer into LDS are the wave-relative LDS byte
// offset (flat aperture rule), which is exactly what VDST expects.
// -----------------------------------------------------------------------------
__device__ __forceinline__ void async_copy_b128(void* lds_dst, const void* gsrc) {
  unsigned int l = (unsigned int)(unsigned long long)lds_dst;
  asm volatile("global_load_async_to_lds_b128 %0, %1, off"
               :: "v"(l), "v"(gsrc) : "memory");
}
__device__ __forceinline__ void async_wait0() {
  asm volatile("s_wait_asynccnt 0x0" ::: "memory");
}

// -----------------------------------------------------------------------------
// One 16x16 WMMA tile of  C[m][n] += dot(A[m,:], W[n,:])  over K (A,W row-major,
// i.e. "A @ W.T").  Uses V_WMMA_F32_16X16X4_F32 so semantics are exact f32.
// A-matrix layout: lane l holds A[l&15][kbase], A[l&15][kbase+1], kbase=2*(l>>4).
// B-matrix layout is the K<->N transpose of that, which for W row-major means
// lane l reads W[l&15][kbase], W[l&15][kbase+1] -- contiguous 8B loads.
// -----------------------------------------------------------------------------
__device__ __forceinline__ v8f wmma_nt(const float* __restrict__ A, int lda,
                                       const float* __restrict__ W, int ldw,
                                       int K, v8f acc) {
  const int lane = threadIdx.x & 31;
  const int half = lane >> 4;
  const int l15  = lane & 15;
  const float* ar = A + (long)l15 * lda + half * 2;
  const float* wr = W + (long)l15 * ldw + half * 2;
#pragma unroll 4
  for (int k = 0; k < K; k += 4) {
    v2f a = *(const v2f*)(ar + k);
    v2f b = *(const v2f*)(wr + k);
    acc = __builtin_amdgcn_wmma_f32_16x16x4_f32(false, a, false, b,
                                                (short)0, acc, false, false);
  }
  return acc;
}

// -------------------- codebook row norms --------------------
__global__ __launch_bounds__(256)
void k_cnorm(const float* __restrict__ cb, float* __restrict__ cnorm) {
  const int i = threadIdx.x;           // 256 codes, 256 threads
  const float* r = cb + (long)i * D_;
  float s = 0.f;
  for (int k = 0; k < D_; ++k) s += r[k] * r[k];
  cnorm[i] = s;
}

// -------------------- quantize: WMMA distances + argmin + gather + vq partial
// 16 feature rows per block, 8 waves; wave w handles code tiles w and w+8.
// score = ||c||^2 - 2*dot  (||x||^2 is row-constant -> irrelevant to argmin).
__global__ __launch_bounds__(256)
void k_quant(const float* __restrict__ feat, const float* __restrict__ cb,
             const float* __restrict__ cnorm,
             float* __restrict__ quant, float* __restrict__ idxOut,
             float* __restrict__ vqPartial) {
  __shared__ float fLds[16 * D_];      // 32 KB feature tile
  __shared__ float redV[8][16];
  __shared__ int   redI[8][16];
  __shared__ int   idxLds[16];
  __shared__ float rs[256];

  const int tid  = threadIdx.x;
  const int wave = tid >> 5;
  const int lane = tid & 31;
  const int half = lane >> 4, l15 = lane & 15;
  const long rowBase = (long)blockIdx.x * 16;

  // async-stage the 32 KB A tile: 256 threads x b128 x 8 passes
  for (int e = tid * 4; e < 16 * D_; e += 256 * 4)
    async_copy_b128(&fLds[e], &feat[rowBase * D_ + e]);
  async_wait0();
  __syncthreads();

  float minv[8]; int mini[8];
#pragma unroll
  for (int v = 0; v < 8; ++v) { minv[v] = 3.4e38f; mini[v] = 0; }

  for (int jj = 0; jj < 2; ++jj) {
    const int j = wave + jj * 8;       // code tile (n in [16j,16j+16))
    v8f acc = {};
    acc = wmma_nt(fLds, D_, cb + (long)(j * 16) * D_, D_, D_, acc);
    const float cn = cnorm[j * 16 + l15];
    const int   n  = j * 16 + l15;
#pragma unroll
    for (int v = 0; v < 8; ++v) {
      const float sc = cn - 2.0f * acc[v];
      if (sc < minv[v] || (sc == minv[v] && n < mini[v])) { minv[v] = sc; mini[v] = n; }
    }
  }
  // reduce across the 16 lanes of each half (masks 1,2,4,8 stay intra-half)
#pragma unroll
  for (int m = 1; m < 16; m <<= 1) {
#pragma unroll
    for (int v = 0; v < 8; ++v) {
      const float ov = __shfl_xor(minv[v], m, 32);
      const int   oi = __shfl_xor(mini[v], m, 32);
      if (ov < minv[v] || (ov == minv[v] && oi < mini[v])) { minv[v] = ov; mini[v] = oi; }
    }
  }
  if (l15 == 0) {                      // lanes 0 & 16 own rows v and v+8
#pragma unroll
    for (int v = 0; v < 8; ++v) {
      redV[wave][v + 8 * half] = minv[v];
      redI[wave][v + 8 * half] = mini[v];
    }
  }
  __syncthreads();
  if (tid < 16) {                      // combine the 8 waves' disjoint n-ranges
    float bv = redV[0][tid]; int bi = redI[0][tid];
    for (int w = 1; w < 8; ++w) {
      const float ov = redV[w][tid]; const int oi = redI[w][tid];
      if (ov < bv || (ov == bv && oi < bi)) { bv = ov; bi = oi; }
    }
    idxLds[tid] = bi;
    idxOut[rowBase + tid] = (float)bi;
  }
  __syncthreads();
  // gather codebook rows -> quantized output; accumulate (f - q)^2
  float s = 0.f;
  for (int e = tid; e < 16 * D_; e += 256) {
    const int r = e >> 9, c = e & (D_ - 1);
    const float q = cb[(long)idxLds[r] * D_ + c];
    const float f = fLds[e];
    quant[rowBase * D_ + e] = q;
    const float d = f - q; s += d * d;
  }
  rs[tid] = s; __syncthreads();
  for (int st = 128; st > 0; st >>= 1) { if (tid < st) rs[tid] += rs[tid + st]; __syncthreads(); }
  if (tid == 0) vqPartial[blockIdx.x] = rs[0];
}

// -------------------- xg = quantized @ w_ih.T + b_ih  ([32768 x 1536]) ------
// A tile async-staged in LDS once per block; 8 waves reuse it (8x less global
// A traffic than direct loads) and hit ds_load_2addr_b64 in the K-loop.
__global__ __launch_bounds__(256)
void k_xg(const float* __restrict__ quant, const float* __restrict__ w_ih,
          const float* __restrict__ b_ih, float* __restrict__ xg) {
  __shared__ float aLds[16 * D_];      // 32 KB
  const int tid  = threadIdx.x;
  const int wave = tid >> 5;
  const int lane = tid & 31;
  const int half = lane >> 4, l15 = lane & 15;
  const long row0 = (long)blockIdx.x * 16;
  const int  n0   = (blockIdx.y * 8 + wave) * 16;

  for (int e = tid * 4; e < 16 * D_; e += 256 * 4)
    async_copy_b128(&aLds[e], &quant[row0 * D_ + e]);
  async_wait0();
  __syncthreads();

  v8f acc = {};
  acc = wmma_nt(aLds, D_, w_ih + (long)n0 * D_, D_, D_, acc);
  const int   n  = n0 + l15;
  const float bb = b_ih[n];
#pragma unroll
  for (int v = 0; v < 8; ++v) {
    const long row = row0 + v + 8 * half;
    xg[row * G3 + n] = acc[v] + bb;
  }
}

// -------------------- sequential GRU: one persistent 1024-thread workgroup --
// h[64x512] f32 lives in LDS (128 KB of the 320 KB WGP pool); w_hh (3 MB)
// streams from L2 each step. Each wave owns 4 (mt,dt) tiles and computes all
// three gates for the same d-columns, so r never needs cross-wave exchange.
__global__ __launch_bounds__(1024)
void k_gru(const float* __restrict__ xg, const float* __restrict__ w_hh,
           const float* __restrict__ b_hh, float* __restrict__ hidden) {
  __shared__ float h[B_ * D_];         // 128 KB
  const int tid  = threadIdx.x;
  const int wave = tid >> 5;
  const int lane = tid & 31;
  const int half = lane >> 4, l15 = lane & 15;

  for (int e = tid; e < B_ * D_; e += 1024) h[e] = 0.f;
  __syncthreads();

  float hn_reg[4][8];
  for (int t = 0; t < T_; ++t) {
#pragma unroll
    for (int i = 0; i < 4; ++i) {
      const int id = wave * 4 + i;     // 128 tiles = 4 Mtiles x 32 Dtiles
      const int mt = id >> 5;
      const int dt = id & 31;
      const float* A = h + mt * 16 * D_;
      v8f aR = {}, aZ = {}, aN = {};
      aR = wmma_nt(A, D_, w_hh + (long)(0 * D_ + dt * 16) * D_, D_, D_, aR);
      aZ = wmma_nt(A, D_, w_hh + (long)(1 * D_ + dt * 16) * D_, D_, D_, aZ);
      aN = wmma_nt(A, D_, w_hh + (long)(2 * D_ + dt * 16) * D_, D_, D_, aN);
      const int d = dt * 16 + l15;
      const float bR = b_hh[d], bZ = b_hh[D_ + d], bN = b_hh[2 * D_ + d];
#pragma unroll
      for (int v = 0; v < 8; ++v) {
        const int m = mt * 16 + v + 8 * half;
        const float* xr = xg + ((long)m * T_ + t) * G3;
        const float r  = 1.f / (1.f + __expf(-(xr[d]        + aR[v] + bR)));
        const float z  = 1.f / (1.f + __expf(-(xr[D_ + d]   + aZ[v] + bZ)));
        const float nn = tanhf(xr[2 * D_ + d] + r * (aN[v] + bN));
        const float ho = h[m * D_ + d];
        hn_reg[i][v] = (1.f - z) * nn + z * ho;
      }
    }
    __syncthreads();                   // all reads of h(t) done
#pragma unroll
    for (int i = 0; i < 4; ++i) {
      const int id = wave * 4 + i;
      const int mt = id >> 5, dt = id & 31;
      const int d  = dt * 16 + l15;
#pragma unroll
      for (int v = 0; v < 8; ++v) {
        const int m = mt * 16 + v + 8 * half;
        h[m * D_ + d] = hn_reg[i][v];
        hidden[((long)m * T_ + t) * D_ + d] = hn_reg[i][v];
      }
    }
    __syncthreads();                   // h(t+1) visible
  }
}

// -------------------- projection GEMM with fused cp-loss epilogue -----------
__global__ __launch_bounds__(256)
void k_proj(const float* __restrict__ hidden, const float* __restrict__ proj_w,
            const float* __restrict__ proj_b, const float* __restrict__ feat,
            float* __restrict__ p1, float* __restrict__ p2) {
  __shared__ float aLds[16 * D_];      // 32 KB
  __shared__ float rs1[256], rs2[256];
  const int tid  = threadIdx.x;
  const int wave = tid >> 5;
  const int lane = tid & 31;
  const int half = lane >> 4, l15 = lane & 15;
  const long row0 = (long)blockIdx.x * 16;
  const int  nt   = blockIdx.y * 8 + wave;
  const int  d    = nt * 16 + l15;

  for (int e = tid * 4; e < 16 * D_; e += 256 * 4)
    async_copy_b128(&aLds[e], &hidden[row0 * D_ + e]);
  async_wait0();
  __syncthreads();

  v8f acc = {};
  acc = wmma_nt(aLds, D_, proj_w + (long)(nt * 16) * D_, D_, D_, acc);
  const float pb = proj_b[d];
  float s1 = 0.f, s2 = 0.f;
#pragma unroll
  for (int v = 0; v < 8; ++v) {
    const long row = row0 + v + 8 * half;
    const int  b   = (int)(row >> 9);
    const int  t   = (int)(row & (T_ - 1));
    const float cp = acc[v] + pb;
    if (t < T_ - 1) { const float df = cp - feat[((long)b * T_ + t + 1) * D_ + d]; s1 += df * df; }
    if (t < T_ - 2) { const float df = cp - feat[((long)b * T_ + t + 2) * D_ + d]; s2 += df * df; }
  }
  rs1[tid] = s1; rs2[tid] = s2; __syncthreads();
  for (int st = 128; st > 0; st >>= 1) {
    if (tid < st) { rs1[tid] += rs1[tid + st]; rs2[tid] += rs2[tid + st]; }
    __syncthreads();
  }
  if (tid == 0) {
    const int bi = blockIdx.y * gridDim.x + blockIdx.x;
    p1[bi] = rs1[0]; p2[bi] = rs2[0];
  }
}

// -------------------- deterministic final reduction + loss scalars ----------
__global__ __launch_bounds__(256)
void k_final(const float* __restrict__ vqP, int nVq,
             const float* __restrict__ p1, const float* __restrict__ p2, int nP,
             float* __restrict__ lossOut) {
  __shared__ float a[256], b[256], c[256];
  const int tid = threadIdx.x;
  float sv = 0.f, s1 = 0.f, s2 = 0.f;
  for (int i = tid; i < nVq; i += 256) sv += vqP[i];
  for (int i = tid; i < nP;  i += 256) { s1 += p1[i]; s2 += p2[i]; }
  a[tid] = sv; b[tid] = s1; c[tid] = s2; __syncthreads();
  for (int st = 128; st > 0; st >>= 1) {
    if (tid < st) { a[tid] += a[tid + st]; b[tid] += b[tid + st]; c[tid] += c[tid + st]; }
    __syncthreads();
  }
  if (tid == 0) {
    const float vq = 1.25f * a[0] / (float)((long)B_ * T_ * D_);   // commit + 0.25*codebook
    const float m1 = b[0] / (float)((long)B_ * (T_ - 1) * D_);
    const float m2 = c[0] / (float)((long)B_ * (T_ - 2) * D_);
    const float cp = 0.5f * (m1 + m2);
    lossOut[0] = cp + vq;   // total
    lossOut[1] = cp;
    lossOut[2] = vq;
  }
}

extern "C" void kernel_launch(void* const* d_in, const int* in_sizes, int n_in,
                              void* d_out, int out_size, void* d_ws, size_t ws_size,
                              hipStream_t stream) {
  const float* feat = (const float*)d_in[0];
  const float* cb   = (const float*)d_in[1];
  const float* w_ih = (const float*)d_in[2];
  const float* w_hh = (const float*)d_in[3];
  const float* b_ih = (const float*)d_in[4];
  const float* b_hh = (const float*)d_in[5];
  const float* pw   = (const float*)d_in[6];
  const float* pb   = (const float*)d_in[7];

  float* out     = (float*)d_out;
  float* quant   = out;                              // B*T*D
  float* idxOut  = out + (long)B_ * T_ * D_;         // B*T (indices as f32)
  float* lossOut = idxOut + (long)B_ * T_;           // total, cp, vq

  float* ws     = (float*)d_ws;
  float* xg     = ws;                                // B*T*3D  = 50,331,648 f32
  float* hidden = xg + (long)B_ * T_ * G3;           // B*T*D   = 16,777,216 f32
  float* cnorm  = hidden + (long)B_ * T_ * D_;       // 256
  float* vqP    = cnorm + K_;                        // 2048
  float* p1     = vqP + 2048;                        // 8192
  float* p2     = p1 + 8192;                         // 8192

  k_cnorm<<<1, 256, 0, stream>>>(cb, cnorm);
  k_quant<<<2048, 256, 0, stream>>>(feat, cb, cnorm, quant, idxOut, vqP);
  k_xg   <<<dim3(2048, 12), 256, 0, stream>>>(quant, w_ih, b_ih, xg);
  k_gru  <<<1, 1024, 0, stream>>>(xg, w_hh, b_hh, hidden);
  k_proj <<<dim3(2048, 4), 256, 0, stream>>>(hidden, pw, pb, feat, p1, p2);
  k_final<<<1, 256, 0, stream>>>(vqP, 2048, p1, p2, 8192, lossOut);
}